// GRUCell_8151847928332
// MI455X (gfx1250) — compile-verified
//
#include <hip/hip_runtime.h>

typedef __attribute__((ext_vector_type(2))) float v2f;
typedef __attribute__((ext_vector_type(4))) float v4f;
typedef __attribute__((ext_vector_type(8))) float v8f;

#define GRU_B (1 << 20)
#define GRU_H 64
#define WG_SIZE 256
#define ROWS_PER_WG 128
#define NUM_RBLOCKS (GRU_B / ROWS_PER_WG) /* 8192 */
#define GRID_DIM 2048
#define LDS_W_FLOATS (6 * 4096)   /* 24576 floats: shuffled weights            */
#define LDS_SCR_FLOATS (8 * 2048) /* per wave: 1024 scr + 1024 hx tile         */
#define LDS_BYTES ((LDS_W_FLOATS + LDS_SCR_FLOATS) * 4) /* 163840 B = 160KB */

#define LOG2E 1.4426950408889634f

// sigmoid(acc + b) with bs = -log2(e)*b  (bias folded into the fma)
__device__ __forceinline__ float sigmoid_pre(float acc, float bs) {
  return __builtin_amdgcn_rcpf(1.0f + __builtin_amdgcn_exp2f(fmaf(-LOG2E, acc, bs)));
}
// tanh(acc + b) with bs = 2*log2(e)*b
__device__ __forceinline__ float tanh_pre(float acc, float bs) {
  float e = __builtin_amdgcn_exp2f(fmaf(2.0f * LOG2E, acc, bs));
  return 1.0f - 2.0f * __builtin_amdgcn_rcpf(e + 1.0f);
}

// Weight LDS layout: idx(k,n) = (k>>1)*128 + (n>>4)*32 + (n&15)*2 + (k&1)
// -> one contiguous ds_load_b64 per B fragment (lane = N, K-pair per lane).
__device__ __forceinline__ v2f bload(const float* wlds, int kc, int nt, int laneLo,
                                     int laneHi) {
  return *(const v2f*)(wlds + (2 * kc + laneHi) * 128 + nt * 32 + laneLo * 2);
}

#define WMMA_F32(A, Bf, C)                                                              \
  __builtin_amdgcn_wmma_f32_16x16x4_f32(false, (A), false, (Bf), (short)0, (C), false,  \
                                        false)

// ---- GEMM cores: acc[nt] (+)= A(16x64) x W(64x64), software-pipelined ----

// A fragments held in registers (x tile). ZERO: first GEMM of a gate, C=inline 0.
template <bool ZERO>
__device__ __forceinline__ void gemm_regA(v8f acc[4], const v2f a[16],
                                          const float* wlds, int laneLo, int laneHi) {
  v2f b[2][4];
#pragma unroll
  for (int nt = 0; nt < 4; ++nt) b[0][nt] = bload(wlds, 0, nt, laneLo, laneHi);
#pragma unroll
  for (int kc = 0; kc < 16; ++kc) {
    const int cur = kc & 1, nxt = cur ^ 1;
    if (kc < 15) {
#pragma unroll
      for (int nt = 0; nt < 4; ++nt) b[nxt][nt] = bload(wlds, kc + 1, nt, laneLo, laneHi);
    }
    if (ZERO && kc == 0) {
#pragma unroll
      for (int nt = 0; nt < 4; ++nt) acc[nt] = WMMA_F32(a[0], b[0][nt], (v8f){});
    } else {
#pragma unroll
      for (int nt = 0; nt < 4; ++nt) acc[nt] = WMMA_F32(a[kc], b[cur][nt], acc[nt]);
    }
  }
}

// A fragments streamed per-kc from a swizzled LDS tile (hx tile)
__device__ __forceinline__ void gemm_ldsA(v8f acc[4], const float* tile, int fidx0,
                                          const float* wlds, int laneLo, int laneHi) {
  v2f a[2], b[2][4];
  a[0] = *(const v2f*)(tile + fidx0);
#pragma unroll
  for (int nt = 0; nt < 4; ++nt) b[0][nt] = bload(wlds, 0, nt, laneLo, laneHi);
#pragma unroll
  for (int kc = 0; kc < 16; ++kc) {
    const int cur = kc & 1, nxt = cur ^ 1;
    if (kc < 15) {
      a[nxt] = *(const v2f*)(tile + (fidx0 ^ (4 * (kc + 1))));
#pragma unroll
      for (int nt = 0; nt < 4; ++nt) b[nxt][nt] = bload(wlds, kc + 1, nt, laneLo, laneHi);
    }
#pragma unroll
    for (int nt = 0; nt < 4; ++nt) acc[nt] = WMMA_F32(a[cur], b[cur][nt], acc[nt]);
  }
}

// A = (r from scratch layout) * (hx from swizzled tile), formed per-kc
__device__ __forceinline__ void gemm_rh(v8f acc[4], const float* scr, const float* tile,
                                        int fidx0, const float* wlds, int laneLo,
                                        int laneHi) {
  v2f r[2], h[2], b[2][4];
  r[0] = *(const v2f*)(scr + laneHi * 32 + laneLo * 2);
  h[0] = *(const v2f*)(tile + fidx0);
#pragma unroll
  for (int nt = 0; nt < 4; ++nt) b[0][nt] = bload(wlds, 0, nt, laneLo, laneHi);
#pragma unroll
  for (int kc = 0; kc < 16; ++kc) {
    const int cur = kc & 1, nxt = cur ^ 1;
    if (kc < 15) {
      r[nxt] = *(const v2f*)(scr + (2 * (kc + 1) + laneHi) * 32 + laneLo * 2);
      h[nxt] = *(const v2f*)(tile + (fidx0 ^ (4 * (kc + 1))));
#pragma unroll
      for (int nt = 0; nt < 4; ++nt) b[nxt][nt] = bload(wlds, kc + 1, nt, laneLo, laneHi);
    }
    v2f a = r[cur] * h[cur];
#pragma unroll
    for (int nt = 0; nt < 4; ++nt) acc[nt] = WMMA_F32(a, b[cur][nt], acc[nt]);
  }
}

// Async-copy one 16x64 f32 tile into LDS, XOR-swizzled row-major:
//   float_idx(m, c) = 64*m + (c ^ 4*m)   (bank-conflict-free frag reads)
__device__ __forceinline__ void async_stage_tile(const float* src, int row0,
                                                 unsigned regionByte, int lane) {
#pragma unroll
  for (int i = 0; i < 8; ++i) {
    int q = i * 32 + lane; // 16B chunk id: row = q>>4, chunk col = q&15
    int row = q >> 4;
    int c4 = q & 15;
    const void* g = (const void*)(src + (row0 + row) * 64 + c4 * 4);
    unsigned l = regionByte + (unsigned)(row * 256) +
                 ((unsigned)(c4 * 16) ^ (unsigned)(row * 16));
    asm volatile("global_load_async_to_lds_b128 %0, %1, off" ::"v"(l), "v"(g)
                 : "memory");
  }
}

__global__ __launch_bounds__(WG_SIZE)
__attribute__((amdgpu_waves_per_eu(4))) void gru_cell_wmma(
    const float* __restrict__ x, const float* __restrict__ hx,
    const float* __restrict__ w_ir, const float* __restrict__ w_hr,
    const float* __restrict__ w_iz, const float* __restrict__ w_hz,
    const float* __restrict__ w_in, const float* __restrict__ w_hn,
    const float* __restrict__ b_r, const float* __restrict__ b_z,
    const float* __restrict__ b_n, float* __restrict__ out) {
  extern __shared__ float lds[];

  const int tid = threadIdx.x;
  const int wave = tid >> 5;
  const int lane = tid & 31;
  const int laneLo = lane & 15;
  const int laneHi = lane >> 4;

  // ---------------- Stage 6 weight matrices into LDS (shuffled layout) ----
  const float* Ws[6] = {w_ir, w_hr, w_iz, w_hz, w_in, w_hn};
#pragma unroll
  for (int w = 0; w < 6; ++w) {
    const float* W = Ws[w];
    float* dst = lds + w * 4096;
#pragma unroll
    for (int i = 0; i < 4; ++i) {
      int e = (tid + i * WG_SIZE) * 4; // 4 consecutive n of one row k
      int k = e >> 6;
      int n = e & 63;
      float4 val = *(const float4*)(W + e);
      int base = (k >> 1) * 128 + (n >> 4) * 32 + (k & 1);
      dst[base + (n & 15) * 2 + 0] = val.x;
      dst[base + ((n + 1) & 15) * 2] = val.y;
      dst[base + ((n + 2) & 15) * 2] = val.z;
      dst[base + ((n + 3) & 15) * 2] = val.w;
    }
  }
  __syncthreads();

  float* scr = lds + LDS_W_FLOATS + wave * 2048; // x-stage -> r -> h-out
  float* hxR = scr + 1024;                       // hx tile, resident all iter
  const unsigned scrByte = (unsigned)(unsigned long long)(const void*)scr;
  const unsigned hxByte = scrByte + 4096u;

  // Pre-scaled biases (folded into activation fma), D layout per N tile
  float brs[4], bzs[4], bns[4];
#pragma unroll
  for (int t = 0; t < 4; ++t) {
    brs[t] = -LOG2E * b_r[t * 16 + laneLo];
    bzs[t] = -LOG2E * b_z[t * 16 + laneLo];
    bns[t] = 2.0f * LOG2E * b_n[t * 16 + laneLo];
  }

  // Swizzled float-index base for A-fragment reads (lane = M, K-pair/lane)
  const int fidx0 = laneLo * 64 + laneLo * 4 + laneHi * 2;

  for (int blk = blockIdx.x; blk < NUM_RBLOCKS; blk += GRID_DIM) {
    const int row0 = blk * ROWS_PER_WG + wave * 16;

    // ---- both tiles HBM -> LDS (async b128, coalesced), single wait ------
    asm volatile("s_wait_dscnt 0x0" ::: "memory"); // regions free of DS readers
    async_stage_tile(x, row0, scrByte, lane);
    async_stage_tile(hx, row0, hxByte, lane);
    asm volatile("s_wait_asynccnt 0x0" ::: "memory");

    // x A-fragments into registers (used by 3 GEMMs; scr is reused for r)
    v2f xA[16];
#pragma unroll
    for (int kc = 0; kc < 16; ++kc)
      xA[kc] = *(const v2f*)(scr + (fidx0 ^ (4 * kc)));
    __builtin_amdgcn_sched_barrier(0);

    v8f acc[4];

    // ---- r = sigmoid(x@w_ir + hx@w_hr + b_r) -----------------------------
    gemm_regA<true>(acc, xA, lds + 0 * 4096, laneLo, laneHi);
    gemm_ldsA(acc, hxR, fidx0, lds + 1 * 4096, laneLo, laneHi);

    // store r to scr in A-readable layout: idx(m,n)=(n>>1)*32+m*2+(n&1)
    // (same-wave DS is in-order: safely follows the xA staging reads)
#pragma unroll
    for (int t = 0; t < 4; ++t) {
      int nb = (8 * t + (laneLo >> 1)) * 32 + (laneLo & 1);
#pragma unroll
      for (int v = 0; v < 8; ++v)
        scr[nb + (8 * laneHi + v) * 2] = sigmoid_pre(acc[t][v], brs[t]);
    }
    __builtin_amdgcn_sched_barrier(0);

    // ---- z = sigmoid(x@w_iz + hx@w_hz + b_z) -----------------------------
    gemm_regA<true>(acc, xA, lds + 2 * 4096, laneLo, laneHi);
    gemm_ldsA(acc, hxR, fidx0, lds + 3 * 4096, laneLo, laneHi);
    v8f zD[4];
#pragma unroll
    for (int t = 0; t < 4; ++t)
#pragma unroll
      for (int v = 0; v < 8; ++v) zD[t][v] = sigmoid_pre(acc[t][v], bzs[t]);
    __builtin_amdgcn_sched_barrier(0);

    // ---- n = tanh(x@w_in + (r*hx)@w_hn + b_n) ----------------------------
    gemm_regA<true>(acc, xA, lds + 4 * 4096, laneLo, laneHi);
    gemm_rh(acc, scr, hxR, fidx0, lds + 5 * 4096, laneLo, laneHi);
    __builtin_amdgcn_sched_barrier(0);

    // ---- h = hx + z*(n - hx): stage in scr, then coalesced b128 stores ---
#pragma unroll
    for (int t = 0; t < 4; ++t)
#pragma unroll
      for (int v = 0; v < 8; ++v) {
        int m = 8 * laneHi + v;
        int di = 64 * m + ((16 * t + laneLo) ^ (4 * m));
        float nv = tanh_pre(acc[t][v], bns[t]);
        float hv = hxR[di]; // hx in D layout, read at point of use
        float zv = zD[t][v];
        scr[di] = hv + zv * (nv - hv);
      }
#pragma unroll
    for (int i = 0; i < 8; ++i) {
      int q = i * 32 + lane;
      int row = q >> 4;
      int c4 = q & 15;
      v4f val = *(const v4f*)(scr + 64 * row + ((4 * c4) ^ (4 * row)));
      *(v4f*)(out + (row0 + row) * 64 + c4 * 4) = val;
    }
  }
}

extern "C" void kernel_launch(void* const* d_in, const int* in_sizes, int n_in,
                              void* d_out, int out_size, void* d_ws, size_t ws_size,
                              hipStream_t stream) {
  (void)in_sizes; (void)n_in; (void)out_size; (void)d_ws; (void)ws_size;
  const float* x = (const float*)d_in[0];
  const float* hx = (const float*)d_in[1];
  const float* w_ir = (const float*)d_in[2];
  const float* w_hr = (const float*)d_in[3];
  const float* w_iz = (const float*)d_in[4];
  const float* w_hz = (const float*)d_in[5];
  const float* w_in = (const float*)d_in[6];
  const float* w_hn = (const float*)d_in[7];
  const float* b_r = (const float*)d_in[8];
  const float* b_z = (const float*)d_in[9];
  const float* b_n = (const float*)d_in[10];
  float* out = (float*)d_out;

  hipLaunchKernelGGL(gru_cell_wmma, dim3(GRID_DIM), dim3(WG_SIZE), LDS_BYTES, stream,
                     x, hx, w_ir, w_hr, w_iz, w_hz, w_in, w_hn, b_r, b_z, b_n, out);
}